// ProcessNeurons_47296179863783
// MI455X (gfx1250) — compile-verified
//
#include <hip/hip_runtime.h>
#include <math.h>

// ---------------------------------------------------------------------------
// ProcessNeurons for MI455X (gfx1250, wave32).
// Shapes: B=8, S=2048, K_IN=512, D_MODEL=1024, N_INPUT=2048, N_PROCESS=4096,
//         HIDDEN=4096, K=256 (top-k).
// Pipeline:
//   0. transpose CW -> CWT [2048,4096], w1 -> w1T [2048,4096]
//   1. GEMM1 (fp32 WMMA 16x16x4): gelu(A[b] @ CWT[idx[b]]) with fused
//      column-max over S into ordered-uint scores (exact fp max).
//   2. h[b,j]   = gelu(b1[j] + sum_{i in set(idx[b])} w1T[i,j])
//   3. rel[b,p] = b2[p] + h[b]·w2[p,:]; fscore = max_s * sigmoid(rel)
//   4. exact top-256 by rank counting (ties -> lower index)
//   5. Bsel gather; GEMM2 recomputes activations for selected columns only
//   6. GEMM3: out = sel_pa @ OP[selp]
// Workspace: ~85 MB.
// ---------------------------------------------------------------------------

typedef float v2f __attribute__((ext_vector_type(2)));
typedef float v8f __attribute__((ext_vector_type(8)));
typedef int   v2i __attribute__((ext_vector_type(2)));

#ifndef __has_builtin
#define __has_builtin(x) 0
#endif
// gfx1250 async global->LDS path (ASYNCcnt). Builtin signature (from the
// compiler diagnostic): (v2i addrspace(1)* src, v2i addrspace(3)* dst,
// imm offset, imm cpol). Guarded: falls back to global->VGPR->LDS staging.
#if __has_builtin(__builtin_amdgcn_global_load_async_to_lds_b64) && \
    __has_builtin(__builtin_amdgcn_s_wait_asynccnt)
#define USE_ASYNC_LDS 1
typedef __attribute__((address_space(1))) v2i gv2i;
typedef __attribute__((address_space(3))) v2i lv2i;
#else
#define USE_ASYNC_LDS 0
#endif

#define BATCH     8
#define SEQ       2048
#define KIN       512
#define DMODEL    1024
#define NINPUT    2048
#define NPROC     4096
#define HIDDEN    4096
#define TOPK      256

__device__ __forceinline__ float gelu_exact(float x) {
    return 0.5f * x * (1.0f + erff(x * 0.70710678118654752440f));
}
// Monotone float->uint mapping so unsigned atomicMax == float max (exact).
__device__ __forceinline__ unsigned ford(float f) {
    unsigned u = __float_as_uint(f);
    return (u & 0x80000000u) ? ~u : (u | 0x80000000u);
}
__device__ __forceinline__ float ford_dec(unsigned u) {
    return (u & 0x80000000u) ? __uint_as_float(u ^ 0x80000000u)
                             : __uint_as_float(~u);
}

// ---------------------------------------------------------------------------
// Generic 32x32-LDS-tiled transpose: src[R][C] -> dst[C][R]
// ---------------------------------------------------------------------------
__global__ __launch_bounds__(256) void transpose_k(const float* __restrict__ src,
                                                   float* __restrict__ dst,
                                                   int R, int C) {
    __shared__ float tile[32][33];
    int c0 = blockIdx.x * 32, r0 = blockIdx.y * 32;
    int tx = threadIdx.x & 31, ty = threadIdx.x >> 5;   // 8 waves of 32
    for (int i = ty; i < 32; i += 8)
        tile[i][tx] = src[(long)(r0 + i) * C + (c0 + tx)];
    __syncthreads();
    for (int i = ty; i < 32; i += 8)
        dst[(long)(c0 + i) * R + (r0 + tx)] = tile[tx][i];
}

// ---------------------------------------------------------------------------
// WMMA fp32 GEMM: C[M x N] = op( A[M x K] @ B[K x N] )
//   - optional row-gather for B (browmap[k] -> physical row of B)
//   - GELU epilogue (compile-time)
//   - MODE 0: store C;  MODE 1: fused column-max into ordered-uint scores
//
// Workgroup 256 threads (8 waves). Tile: 128(M) x 64(N), Kc = 32.
// Wave w: M-block (w>>1)*32, N-block (w&1)*32 -> four 16x16 accumulators.
//
// Fragment-ready LDS layouts (everything is one aligned ds_load_b64):
//   As[m][k]          : row stride 34 floats (136B, 8B-aligned rows);
//                       A frag lane pair = {A[m][k], A[m][k+1]} contiguous.
//   Bs2[k/2][n*2+k&1] : k-pair interleaved, pair stride 2*TN+32 floats so the
//                       two half-waves (adjacent k-pairs) use disjoint banks;
//                       B frag lane pair = {B[k][n], B[k+1][n]} contiguous.
//
// A-tile staging uses GLOBAL_LOAD_ASYNC_TO_LDS_B64 (ASYNCcnt, no VGPR
// round-trip); B-tile staging stays manual (interleaved scatter). Async
// writes fenced with s_wait_asynccnt 0 before the workgroup barrier.
//
// fp32 WMMA layouts per CDNA5 ISA 7.12.2:
//   A 16x4 : v0 = K{0|2}, v1 = K{1|3}; M = lane&15 (both halves)
//   B 4x16 : v0 = row K{0|2}, v1 = row K{1|3}; N = lane&15
//   C 16x16: vgpr i -> M = i + 8*(lane>=16), N = lane&15
// ---------------------------------------------------------------------------
template <bool GELU, int MODE>
__global__ __launch_bounds__(256) void wmma_gemm_k(
    const float* __restrict__ A, int lda, long strideA,
    const float* __restrict__ B, int ldb, long strideB,
    const int*  __restrict__ browmap, int strideMap,
    int Kdim,
    float* __restrict__ C, int ldc, long strideC,
    unsigned* __restrict__ scores, int strideScores) {

    constexpr int TM = 128, TN = 64, KC = 32;
    constexpr int AP = KC + 2;        // 34 floats: 8B-aligned rows, odd bank step
    constexpr int BP = 2 * TN + 32;   // 160 floats: half-waves on disjoint banks
    __shared__ float As[TM][AP];
    __shared__ float Bs2[KC / 2][BP];

    const int bz = blockIdx.z;
    A += (long)bz * strideA;
    B += (long)bz * strideB;
    if (browmap) browmap += (long)bz * strideMap;
    if (MODE == 0) C += (long)bz * strideC;
    else           scores += (long)bz * strideScores;

    const int m0 = blockIdx.y * TM;
    const int n0 = blockIdx.x * TN;
    const int t    = threadIdx.x;
    const int wave = t >> 5;
    const int lane = t & 31;          // wave32 on gfx1250
    const int hf   = lane >> 4;       // half-wave select
    const int l15  = lane & 15;
    const int wm   = (wave >> 1) * 32;
    const int wn   = (wave & 1) * 32;

    v8f c00 = {0.f,0.f,0.f,0.f,0.f,0.f,0.f,0.f};
    v8f c01 = c00, c10 = c00, c11 = c00;

    for (int k0 = 0; k0 < Kdim; k0 += KC) {
        __syncthreads();   // previous tile fully consumed by all waves
        {   // Stage A tile: 128 rows x 32 cols, 16 floats / thread.
            int r = t >> 1, cb = (t & 1) * 16;
            const float* srcA = A + (long)(m0 + r) * lda + (k0 + cb);
            if (k0 + KC < Kdim) __builtin_prefetch(srcA + KC, 0, 1); // global_prefetch_b8
#if USE_ASYNC_LDS
            gv2i* gsrc = (gv2i*)const_cast<float*>(srcA);  // 8B-aligned
            lv2i* ldst = (lv2i*)&As[r][cb];                // 8B-aligned
#pragma unroll
            for (int i = 0; i < 8; ++i)
                __builtin_amdgcn_global_load_async_to_lds_b64(
                    gsrc + i, ldst + i, 0, 0);
#else
            const float4* s4 = (const float4*)srcA;
#pragma unroll
            for (int i = 0; i < 4; ++i) {
                float4 v = s4[i];
                As[r][cb + 4*i + 0] = v.x;
                As[r][cb + 4*i + 1] = v.y;
                As[r][cb + 4*i + 2] = v.z;
                As[r][cb + 4*i + 3] = v.w;
            }
#endif
        }
        {   // Stage B tile: 32 (gathered) rows x 64 cols, k-pair interleaved.
            int kr = t >> 3, cb = (t & 7) * 8;
            int grow = browmap ? browmap[k0 + kr] : (k0 + kr);
            const float* srcB = B + (long)grow * ldb + (n0 + cb);
            float* dst = &Bs2[kr >> 1][(kr & 1)];
#pragma unroll
            for (int i = 0; i < 8; ++i) dst[(cb + i) * 2] = srcB[i];
        }
#if USE_ASYNC_LDS
        __builtin_amdgcn_s_wait_asynccnt(0);   // async LDS writes visible
#endif
        __syncthreads();

#pragma unroll
        for (int kk = 0; kk < KC; kk += 4) {
            // Each fragment: one aligned 8-byte LDS load.
            v2f a0 = *(const v2f*)&As[wm +      l15][kk + hf * 2];
            v2f a1 = *(const v2f*)&As[wm + 16 + l15][kk + hf * 2];
            v2f b0 = *(const v2f*)&Bs2[(kk >> 1) + hf][(wn +      l15) * 2];
            v2f b1 = *(const v2f*)&Bs2[(kk >> 1) + hf][(wn + 16 + l15) * 2];
            c00 = __builtin_amdgcn_wmma_f32_16x16x4_f32(false, a0, false, b0,
                                                        (short)0, c00, false, false);
            c01 = __builtin_amdgcn_wmma_f32_16x16x4_f32(false, a0, false, b1,
                                                        (short)0, c01, false, false);
            c10 = __builtin_amdgcn_wmma_f32_16x16x4_f32(false, a1, false, b0,
                                                        (short)0, c10, false, false);
            c11 = __builtin_amdgcn_wmma_f32_16x16x4_f32(false, a1, false, b1,
                                                        (short)0, c11, false, false);
        }
    }

    if (MODE == 0) {
        const int col0 = n0 + wn + l15;
#pragma unroll
        for (int i = 0; i < 8; ++i) {
            int r0 = m0 + wm + hf * 8 + i;
            int r1 = r0 + 16;
            float v00 = c00[i], v01 = c01[i], v10 = c10[i], v11 = c11[i];
            if (GELU) {
                v00 = gelu_exact(v00); v01 = gelu_exact(v01);
                v10 = gelu_exact(v10); v11 = gelu_exact(v11);
            }
            C[(long)r0 * ldc + col0]      = v00;
            C[(long)r0 * ldc + col0 + 16] = v01;
            C[(long)r1 * ldc + col0]      = v10;
            C[(long)r1 * ldc + col0 + 16] = v11;
        }
    } else {
        // Column max of gelu(x) over this 32-row block, then global atomicMax.
        float mx0 = -3.4e38f, mx1 = -3.4e38f;
#pragma unroll
        for (int i = 0; i < 8; ++i) {
            mx0 = fmaxf(mx0, fmaxf(GELU ? gelu_exact(c00[i]) : c00[i],
                                   GELU ? gelu_exact(c10[i]) : c10[i]));
            mx1 = fmaxf(mx1, fmaxf(GELU ? gelu_exact(c01[i]) : c01[i],
                                   GELU ? gelu_exact(c11[i]) : c11[i]));
        }
        mx0 = fmaxf(mx0, __shfl_xor(mx0, 16, 32));
        mx1 = fmaxf(mx1, __shfl_xor(mx1, 16, 32));
        if (hf == 0) {
            atomicMax(&scores[n0 + wn + l15],      ford(mx0));
            atomicMax(&scores[n0 + wn + 16 + l15], ford(mx1));
        }
    }
}

// ---------------------------------------------------------------------------
// h[b,j] = gelu(b1[j] + sum over unique selected inputs of w1T[i,j])
// ---------------------------------------------------------------------------
__global__ __launch_bounds__(256) void mlp1_k(const float* __restrict__ w1T,
                                              const float* __restrict__ b1,
                                              const int*  __restrict__ sidx,
                                              float* __restrict__ h) {
    __shared__ unsigned char flags[NINPUT];
    const int b = blockIdx.y;
    const int t = threadIdx.x;
    for (int i = t; i < NINPUT; i += 256) flags[i] = 0;
    __syncthreads();
    for (int u = t; u < KIN; u += 256) flags[sidx[b * KIN + u]] = 1;  // dedupe
    __syncthreads();
    const int j = blockIdx.x * 256 + t;
    float acc = b1[j];
    for (int i = 0; i < NINPUT; ++i)
        if (flags[i]) acc += w1T[(long)i * HIDDEN + j];
    h[b * HIDDEN + j] = gelu_exact(acc);
}

// ---------------------------------------------------------------------------
// fscore[b,p] = decode(scores[b,p]) * sigmoid(b2[p] + h[b]·w2[p,:])
// One wave per output row p -> coalesced float4 streams + wave32 reduction.
// ---------------------------------------------------------------------------
__global__ __launch_bounds__(256) void mlp2_k(const float* __restrict__ h,
                                              const float* __restrict__ w2,
                                              const float* __restrict__ b2,
                                              const unsigned* __restrict__ scores,
                                              float* __restrict__ fscore) {
    __shared__ float hs[HIDDEN];
    const int b = blockIdx.y;
    const int t = threadIdx.x, wave = t >> 5, lane = t & 31;
    for (int i = t; i < HIDDEN; i += 256) hs[i] = h[b * HIDDEN + i];
    __syncthreads();
    for (int rep = 0; rep < 8; ++rep) {
        const int p = blockIdx.x * 64 + wave * 8 + rep;
        const float4* row = (const float4*)(w2 + (long)p * HIDDEN);
        float acc = 0.f;
        for (int j4 = lane; j4 < HIDDEN / 4; j4 += 32) {
            float4 w = row[j4];
            acc += w.x * hs[j4 * 4 + 0] + w.y * hs[j4 * 4 + 1] +
                   w.z * hs[j4 * 4 + 2] + w.w * hs[j4 * 4 + 3];
        }
#pragma unroll
        for (int off = 16; off > 0; off >>= 1) acc += __shfl_xor(acc, off, 32);
        if (lane == 0) {
            float rel = acc + b2[p];
            float sg  = 1.f / (1.f + expf(-rel));
            fscore[b * NPROC + p] = ford_dec(scores[b * NPROC + p]) * sg;
        }
    }
}

// ---------------------------------------------------------------------------
// Exact top-K by rank counting; ties break to lower index (matches lax.top_k).
// Ranks are a permutation -> deterministic rank-ordered compaction.
// ---------------------------------------------------------------------------
__global__ __launch_bounds__(1024) void topk_k(const float* __restrict__ fscore,
                                               int* __restrict__ selp, int K) {
    __shared__ float fs[NPROC];
    const int b = blockIdx.x, t = threadIdx.x;
    for (int i = t; i < NPROC; i += 1024) fs[i] = fscore[b * NPROC + i];
    __syncthreads();
    for (int rep = 0; rep < NPROC / 1024; ++rep) {
        const int p = rep * 1024 + t;
        const float v = fs[p];
        int rank = 0;
        for (int j = 0; j < NPROC; ++j) {          // uniform j -> LDS broadcast
            float u = fs[j];
            rank += (u > v) || (u == v && j < p);
        }
        if (rank < K) selp[b * K + rank] = p;
    }
}

// ---------------------------------------------------------------------------
// Bsel[b][k][r] = CWT[idx[b,k]][selp[b,r]]  (pre-gather selected columns)
// ---------------------------------------------------------------------------
__global__ __launch_bounds__(256) void gatherB_k(const float* __restrict__ CWT,
                                                 const int* __restrict__ sidx,
                                                 const int* __restrict__ selp,
                                                 float* __restrict__ Bsel) {
    const int b  = blockIdx.y;
    const int id = blockIdx.x * 256 + threadIdx.x;  // KIN*TOPK per batch
    const int kk = id >> 8, r = id & 255;
    Bsel[((long)b * KIN + kk) * TOPK + r] =
        CWT[(long)sidx[b * KIN + kk] * NPROC + selp[b * TOPK + r]];
}

// ---------------------------------------------------------------------------
extern "C" void kernel_launch(void* const* d_in, const int* in_sizes, int n_in,
                              void* d_out, int out_size, void* d_ws, size_t ws_size,
                              hipStream_t stream) {
    const float* act  = (const float*)d_in[0];   // [8,2048,512]
    const int*   sidx = (const int*)  d_in[1];   // [8,512]
    /* d_in[2] = k (==256, fixed by setup) */
    const float* cw   = (const float*)d_in[3];   // [4096,2048]
    const float* op   = (const float*)d_in[4];   // [4096,1024]
    const float* w1   = (const float*)d_in[5];   // [4096,2048]
    const float* b1   = (const float*)d_in[6];   // [4096]
    const float* w2   = (const float*)d_in[7];   // [4096,4096]
    const float* b2   = (const float*)d_in[8];   // [4096]
    float* out = (float*)d_out;                  // [8,2048,1024]

    char* ws = (char*)d_ws;                      // ~85 MB used
    float*    CWT    = (float*)   (ws);                             // 32 MB
    float*    w1T    = (float*)   (ws + (32ull << 20));             // 32 MB
    float*    Bsel   = (float*)   (ws + (64ull << 20));             //  4 MB
    float*    selpa  = (float*)   (ws + (68ull << 20));             // 16 MB
    float*    h      = (float*)   (ws + (84ull << 20));             // 128 KB
    unsigned* scr    = (unsigned*)(ws + (84ull << 20) + (1u << 18));// 128 KB
    float*    fsc    = (float*)   (ws + (84ull << 20) + (2u << 18));// 128 KB
    int*      selp   = (int*)     (ws + (84ull << 20) + (3u << 18));//   8 KB

    // 0 is the identity of the ordered-uint max encoding.
    (void)hipMemsetAsync(scr, 0, BATCH * NPROC * sizeof(unsigned), stream);

    transpose_k<<<dim3(NINPUT / 32, NPROC / 32), 256, 0, stream>>>(cw, CWT, NPROC, NINPUT);
    transpose_k<<<dim3(NINPUT / 32, HIDDEN / 32), 256, 0, stream>>>(w1, w1T, HIDDEN, NINPUT);

    // GEMM1: gelu(A @ CWT[idx]) with fused column-max -> scores
    wmma_gemm_k<true, 1><<<dim3(NPROC / 64, SEQ / 128, BATCH), 256, 0, stream>>>(
        act, KIN, (long)SEQ * KIN,
        CWT, NPROC, 0,
        sidx, KIN,
        KIN,
        nullptr, 0, 0,
        scr, NPROC);

    mlp1_k<<<dim3(HIDDEN / 256, BATCH), 256, 0, stream>>>(w1T, b1, sidx, h);
    mlp2_k<<<dim3(NPROC / 64, BATCH), 256, 0, stream>>>(h, w2, b2, scr, fsc);
    topk_k<<<dim3(BATCH), 1024, 0, stream>>>(fsc, selp, TOPK);
    gatherB_k<<<dim3(KIN * TOPK / 256, BATCH), 256, 0, stream>>>(CWT, sidx, selp, Bsel);

    // GEMM2: recompute gelu-activations only for the selected 256 columns
    wmma_gemm_k<true, 0><<<dim3(TOPK / 64, SEQ / 128, BATCH), 256, 0, stream>>>(
        act, KIN, (long)SEQ * KIN,
        Bsel, TOPK, (long)KIN * TOPK,
        nullptr, 0,
        KIN,
        selpa, TOPK, (long)SEQ * TOPK,
        nullptr, 0);

    // GEMM3: out = sel_pa @ OP[selp]
    wmma_gemm_k<false, 0><<<dim3(DMODEL / 64, SEQ / 128, BATCH), 256, 0, stream>>>(
        selpa, TOPK, (long)SEQ * TOPK,
        op, DMODEL, 0,
        selp, TOPK,
        TOPK,
        out, DMODEL, (long)SEQ * DMODEL,
        nullptr, 0);

    (void)in_sizes; (void)n_in; (void)out_size; (void)ws_size;
}